// SSDTableBatchedEmbeddingBags_21509196219244
// MI455X (gfx1250) — compile-verified
//
#include <hip/hip_runtime.h>
#include <hip/hip_bf16.h>
#include <stdint.h>

// SSD table-batched embedding-bag forward (gather + SUM pool), MI455X / gfx1250.
// T=4 tables, B=8192 bags/table, L=20 rows/bag, ROWS=250000, D=128 fp32.
// Bandwidth-bound: ~335MB of random 512B row reads -> wave32-per-bag with
// one global_load_async_to_lds_b128 per lane per row (512B/row coalesced),
// double-buffered through LDS via ASYNCcnt. Row addressing: uniform 64-bit
// base + divergent 32-bit byte offset (1M rows * 512B < 2^29) so the async
// loads can select the SADDR (GVS) form.

namespace {
constexpr int kT = 4;
constexpr int kB = 8192;
constexpr int kL = 20;
constexpr int kRows = 250000;
constexpr int kD = 128;

constexpr int kWavesPerBlock = 8;           // 256 threads = 8 wave32
constexpr int kThreads = kWavesPerBlock * 32;
constexpr int kCH = 4;                      // rows per async chunk
constexpr int kNCH = kL / kCH;              // 5 chunks
static_assert(kL % kCH == 0, "L must be a multiple of chunk size");
static_assert(kT == 4, "bag decode uses &3 / >>2");
}  // namespace

// Async-copy builtin params: pointers to 4-int vectors in AS1 / AS3
// (signature confirmed by the round-1 clang diagnostic).
typedef int v4i __attribute__((ext_vector_type(4)));
typedef __attribute__((address_space(1))) v4i* as1_v4i_p;
typedef __attribute__((address_space(3))) v4i* as3_v4i_p;

// Per-lane 16B async copy: global -> LDS (one wave32 issue moves a full 512B row).
__device__ __forceinline__ void async_copy_b128(const void* gsrc, void* ldst) {
#if __has_builtin(__builtin_amdgcn_global_load_async_to_lds_b128)
  __builtin_amdgcn_global_load_async_to_lds_b128(
      (as1_v4i_p)(uintptr_t)gsrc,
      (as3_v4i_p)(uint32_t)(uintptr_t)ldst,   // low 32 bits of flat addr == LDS offset
      0 /*imm offset*/, 0 /*cpol*/);
#else
  asm volatile("global_load_async_to_lds_b128 %0, %1, off"
               :
               : "v"((uint32_t)(uintptr_t)ldst), "v"((uint64_t)(uintptr_t)gsrc)
               : "memory");
#endif
}

__device__ __forceinline__ void wait_async_le_chunk() {
#if __has_builtin(__builtin_amdgcn_s_wait_asynccnt)
  __builtin_amdgcn_s_wait_asynccnt(kCH);
#else
  asm volatile("s_wait_asynccnt 4" ::: "memory");
#endif
  asm volatile("" ::: "memory");
}

__device__ __forceinline__ void wait_async_zero() {
#if __has_builtin(__builtin_amdgcn_s_wait_asynccnt)
  __builtin_amdgcn_s_wait_asynccnt(0);
#else
  asm volatile("s_wait_asynccnt 0" ::: "memory");
#endif
  asm volatile("" ::: "memory");
}

__global__ __launch_bounds__(kThreads) void emb_bag_pool_kernel(
    const float* __restrict__ weights,   // [T*ROWS, D]
    const int*   __restrict__ indices,   // [T, B, L]
    float*       __restrict__ out) {     // [B, T*D]
  const uint32_t lane = threadIdx.x & 31u;
  const uint32_t wave = threadIdx.x >> 5;
  const uint32_t bag  = blockIdx.x * kWavesPerBlock + wave;   // [0, T*B)
  const uint32_t t = bag & (kT - 1);
  const uint32_t b = bag >> 2;

  // Load this bag's 20 indices and fold them into 32-bit byte offsets for
  // this lane's 16B slice: off = (t*ROWS + idx)*512 + lane*16 < 2^29.
  const int* bag_idx = indices + ((size_t)t * kB + (size_t)b) * kL;
  const uint32_t tbase = t * (uint32_t)kRows;
  const uint32_t laneoff = lane * 16u;
  uint32_t boff[kL];
#pragma unroll
  for (int l = 0; l < kL; ++l)
    boff[l] = (((tbase + (uint32_t)bag_idx[l]) << 9) + laneoff);

  const char* wbase = (const char*)weights;   // uniform 64-bit base -> SADDR form

  // Double-buffered staging: 8 waves x 2 sets x 4 rows x 512B = 32KB/block.
  __shared__ float4 stage[kWavesPerBlock][2][kCH][32];

  float ax = 0.f, ay = 0.f, az = 0.f, aw = 0.f;

  // Prologue: chunk 0 -> set 0.
#pragma unroll
  for (int r = 0; r < kCH; ++r)
    async_copy_b128(wbase + boff[r], &stage[wave][0][r][lane]);

#pragma unroll
  for (int c = 0; c < kNCH; ++c) {
    const int set = c & 1;
    if (c + 1 < kNCH) {
      // Issue next chunk into the other buffer set, then wait for the oldest
      // kCH async loads (chunk c) to land; async loads complete in order.
#pragma unroll
      for (int r = 0; r < kCH; ++r)
        async_copy_b128(wbase + boff[(c + 1) * kCH + r],
                        &stage[wave][set ^ 1][r][lane]);
      wait_async_le_chunk();
    } else {
      wait_async_zero();
    }
    // Accumulate chunk c out of LDS (ds_load_b128 per row).
#pragma unroll
    for (int r = 0; r < kCH; ++r) {
      float4 v = stage[wave][set][r][lane];
      ax += v.x; ay += v.y; az += v.z; aw += v.w;
    }
  }

  // out[b, t*D + lane*4 .. +3]: 32-bit byte offset + uniform base, b128 store.
  const uint32_t ooff = (b * (uint32_t)(kT * kD) + t * (uint32_t)kD) * 4u + laneoff;
  float4* op = (float4*)((char*)out + ooff);
  *op = make_float4(ax, ay, az, aw);
}

extern "C" void kernel_launch(void* const* d_in, const int* in_sizes, int n_in,
                              void* d_out, int out_size, void* d_ws, size_t ws_size,
                              hipStream_t stream) {
  const float* weights = (const float*)d_in[0];   // [T*ROWS, D] fp32
  const int*   indices = (const int*)d_in[1];     // [T, B, L] int
  float*       out     = (float*)d_out;           // [B, T*D] fp32
  (void)in_sizes; (void)n_in; (void)d_ws; (void)ws_size; (void)out_size;

  const int bags = kT * kB;                       // 32768 waves, one per bag
  dim3 grid(bags / kWavesPerBlock);               // 4096 blocks
  dim3 block(kThreads);                           // 256 threads (8 wave32)
  emb_bag_pool_kernel<<<grid, block, 0, stream>>>(weights, indices, out);
}